// _PanoramicReasoningBlockND_58901181497737
// MI455X (gfx1250) — compile-verified
//
#include <hip/hip_runtime.h>

// CDNA5 (gfx1250) fused "panoramic reasoning block", fp32 WMMA path.
// 2 tiles per workgroup: each weight A-operand load feeds 2 WMMAs, halving
// L2 weight traffic. All GEMMs via v_wmma_f32_16x16x4_f32.

typedef float v2f __attribute__((ext_vector_type(2)));
typedef float v8f __attribute__((ext_vector_type(8)));

#define C_INC   1024
#define C_INT   512
#define NPOS    9
#define NP      16
#define HW      48
#define XK      1028   // xT row stride (dwords): %64==4 -> conflict-free b64 reads
#define CK      516    // thT/phT(yT) row stride: %64==4
#define GK      12     // gcm row stride (only k=0..11 read by y-GEMM)
#define FK      20     // f row stride
#define NTHREADS 512
#define NWAVES   16
#define TPB      2     // tiles per block

__device__ __forceinline__ v8f wmma4(v2f a, v2f b, v8f c) {
    return __builtin_amdgcn_wmma_f32_16x16x4_f32(false, a, false, b,
                                                 (short)0, c, false, false);
}

// One projection pass (M=512, N=16 per tile, K=1024) over 2 tiles.
// CMAJOR=true  -> store c-major into gcm  [o*GK + n]
// CMAJOR=false -> store transposed into thT/phT [n*CK + o]
template <bool CMAJOR>
__device__ __forceinline__ void proj_pass(const float* __restrict__ W,
                                          const float* __restrict__ bias,
                                          const float* b0p, const float* b1p,
                                          float* d0, float* d1,
                                          int wave, int l16, int half)
{
    for (int rt = wave; rt < 32; rt += NWAVES) {
        const int rb = rt << 4;
        const float* a = W + (size_t)(rb + l16) * C_INC + 2 * half;
        v8f acc0 = {}, acc1 = {};
        #pragma unroll 4
        for (int k = 0; k < C_INC; k += 4) {
            v2f av = *(const v2f*)(a + k);
            acc0 = wmma4(av, *(const v2f*)(b0p + k), acc0);
            acc1 = wmma4(av, *(const v2f*)(b1p + k), acc1);
        }
        if (l16 < NPOS) {             // keep padded columns exactly zero
            #pragma unroll
            for (int r = 0; r < 8; ++r) {
                const int o = rb + r + 8 * half;
                const float bv = bias[o];
                if (CMAJOR) {
                    d0[o * GK + l16] = acc0[r] + bv;
                    d1[o * GK + l16] = acc1[r] + bv;
                } else {
                    d0[l16 * CK + o] = acc0[r] + bv;
                    d1[l16 * CK + o] = acc1[r] + bv;
                }
            }
        }
    }
}

__global__ __launch_bounds__(NTHREADS, 1)
void panoramic_tile_kernel(const float* __restrict__ x,
                           const float* __restrict__ g_w,  const float* __restrict__ g_b,
                           const float* __restrict__ th_w, const float* __restrict__ th_b,
                           const float* __restrict__ ph_w, const float* __restrict__ ph_b,
                           const float* __restrict__ w_w,  const float* __restrict__ w_b,
                           float* __restrict__ out)
{
    // Rows 0..8 live, row 9 = shared zero row (lanes 9..15 clamp to it).
    __shared__ float xT [TPB][10 * XK];     // x tile, transposed [n][c]
    __shared__ float thT[TPB][10 * CK];     // theta [n][ci]; reused as y after f
    __shared__ float phT[TPB][10 * CK];     // phi   [n][ci]
    __shared__ float gcm[TPB][C_INT * GK];  // g, c-major [ci][m]
    __shared__ float fT [TPB][NP * FK];     // f [n][m]

    const int tid  = threadIdx.x;
    const int wave = tid >> 5;
    const int lane = tid & 31;
    const int half = lane >> 4;                 // A/B operand half (K+0/1 vs K+2/3)
    const int l16  = lane & 15;
    const int lrow = (l16 < NPOS) ? l16 : NPOS; // clamped LDS row for padded lanes

    const int pid = blockIdx.x;                 // 1024 pairs of tiles
    const int t0  = pid << 1;
    const int bb  = t0 >> 8;
    const int hh  = (t0 >> 4) & 15;
    const int ww0 = t0 & 15;                    // even; pair is (ww0, ww0+1)
    const size_t imgbase = (size_t)bb * C_INC * HW * HW;

    // ---- Phase 0: zero clamp rows / padding columns ------------------------
    #pragma unroll
    for (int u = 0; u < TPB; ++u) {
        for (int i = tid; i < XK; i += NTHREADS) xT[u][NPOS * XK + i] = 0.0f;
        for (int i = tid; i < CK; i += NTHREADS) {
            thT[u][NPOS * CK + i] = 0.0f;
            phT[u][NPOS * CK + i] = 0.0f;
        }
        for (int i = tid; i < C_INT * 3; i += NTHREADS)
            gcm[u][(i / 3) * GK + 9 + (i % 3)] = 0.0f;
    }

    // ---- Phase 1: gather both tiles into LDS, transposed [n][c] ------------
    #pragma unroll
    for (int u = 0; u < TPB; ++u) {
        for (int n = 0; n < NPOS; ++n) {
            const int kh = n / 3, kw = n - 3 * (n / 3);
            const float* src = x + imgbase + (size_t)(hh * 3 + kh) * HW
                                 + ((ww0 + u) * 3 + kw);
            for (int c = tid; c < C_INC; c += NTHREADS)
                xT[u][n * XK + c] = src[(size_t)c * (HW * HW)];
        }
    }
    __syncthreads();

    // ---- Phase 2: g/theta/phi projections ----------------------------------
    {
        const float* b0p = &xT[0][lrow * XK + 2 * half];
        const float* b1p = &xT[1][lrow * XK + 2 * half];
        proj_pass<true >(g_w,  g_b,  b0p, b1p, gcm[0], gcm[1], wave, l16, half);
        proj_pass<false>(th_w, th_b, b0p, b1p, thT[0], thT[1], wave, l16, half);
        proj_pass<false>(ph_w, ph_b, b0p, b1p, phT[0], phT[1], wave, l16, half);
    }
    __syncthreads();

    // ---- Phase 3a: f[n][m] = <theta[:,n], phi[:,m]> / 9 (16x16, K=512) -----
    if (wave < TPB) {
        const int u = wave;
        const float* a = &thT[u][lrow * CK + 2 * half];
        const float* b = &phT[u][lrow * CK + 2 * half];
        v8f acc = {};
        #pragma unroll 8
        for (int k = 0; k < C_INT; k += 4)
            acc = wmma4(*(const v2f*)(a + k), *(const v2f*)(b + k), acc);
        const float inv9 = 1.0f / 9.0f;
        #pragma unroll
        for (int r = 0; r < 8; ++r)
            fT[u][(r + 8 * half) * FK + l16] = acc[r] * inv9;
    }
    __syncthreads();

    // ---- Phase 3b: y[c][n] = sum_m f[n][m] g[c][m]; y overwrites thT -------
    for (int it = wave; it < 64; it += NWAVES) {
        const int u  = it & 1;
        const int rb = (it >> 1) << 4;
        const float* a = &gcm[u][(size_t)(rb + l16) * GK + 2 * half];
        const float* b = &fT[u][l16 * FK + 2 * half];
        v8f acc = {};
        #pragma unroll
        for (int k = 0; k < 12; k += 4)
            acc = wmma4(*(const v2f*)(a + k), *(const v2f*)(b + k), acc);
        if (l16 < NPOS) {
            #pragma unroll
            for (int r = 0; r < 8; ++r)
                thT[u][l16 * CK + rb + r + 8 * half] = acc[r];
        }
    }
    __syncthreads();

    // ---- Phase 4: z = W*y + b + xw, scatter back to NCHW -------------------
    for (int rt = wave; rt < 64; rt += NWAVES) {
        const int rb = rt << 4;
        const float* a  = w_w + (size_t)(rb + l16) * C_INT + 2 * half;
        const float* b0 = &thT[0][lrow * CK + 2 * half];   // y, tile 0
        const float* b1 = &thT[1][lrow * CK + 2 * half];   // y, tile 1
        v8f acc0 = {}, acc1 = {};
        #pragma unroll 4
        for (int k = 0; k < C_INT; k += 4) {
            v2f av = *(const v2f*)(a + k);
            acc0 = wmma4(av, *(const v2f*)(b0 + k), acc0);
            acc1 = wmma4(av, *(const v2f*)(b1 + k), acc1);
        }
        if (l16 < NPOS) {
            const int kh = l16 / 3, kw = l16 - 3 * (l16 / 3);
            float* dst0 = out + imgbase + (size_t)(hh * 3 + kh) * HW + (ww0 * 3 + kw);
            float* dst1 = dst0 + 3;                        // (ww0+1)*3 + kw
            #pragma unroll
            for (int r = 0; r < 8; ++r) {
                const int o = rb + r + 8 * half;
                const float bv = w_b[o];
                dst0[(size_t)o * (HW * HW)] = acc0[r] + bv + xT[0][l16 * XK + o];
                dst1[(size_t)o * (HW * HW)] = acc1[r] + bv + xT[1][l16 * XK + o];
            }
        }
    }
}

extern "C" void kernel_launch(void* const* d_in, const int* in_sizes, int n_in,
                              void* d_out, int out_size, void* d_ws, size_t ws_size,
                              hipStream_t stream) {
    const float* x    = (const float*)d_in[0];
    const float* g_w  = (const float*)d_in[1];
    const float* g_b  = (const float*)d_in[2];
    const float* th_w = (const float*)d_in[3];
    const float* th_b = (const float*)d_in[4];
    const float* ph_w = (const float*)d_in[5];
    const float* ph_b = (const float*)d_in[6];
    const float* w_w  = (const float*)d_in[7];
    const float* w_b  = (const float*)d_in[8];
    float* out = (float*)d_out;

    dim3 grid(1024);        // 2048 tiles / 2 per block
    dim3 block(NTHREADS);
    hipLaunchKernelGGL(panoramic_tile_kernel, grid, block, 0, stream,
                       x, g_w, g_b, th_w, th_b, ph_w, ph_b, w_w, w_b, out);
}